// ASTDecoder_22565758173967
// MI455X (gfx1250) — compile-verified
//
#include <hip/hip_runtime.h>
#include <hip/hip_bf16.h>

// GCN on banded graphs, fully fused per-graph in LDS.
// Shapes: BATCH=2048, EMB=256, HID=64, OUT=128, NODES=256, LAYERS=3.
// One workgroup (512 threads = 16 wave32) per graph.
// All GEMMs use V_WMMA_F32_16X16X4_F32 (exact fp32, matches reference).

typedef __attribute__((ext_vector_type(2))) float v2f;
typedef __attribute__((ext_vector_type(8))) float v8f;

#define NODES 256
#define HID   64
#define OUTD  128
#define XS    68    // row stride (floats) for X/H in LDS: 4*row -> bank-conflict-free A frags
#define WSTR  144   // row stride (floats) for W in LDS: 144 % 64 == 16 -> conflict-free B frags

__global__ __launch_bounds__(512) void gcn_fused_kernel(
    const float* __restrict__ emb,    // [2048][256]
    const float* __restrict__ W_emb,  // [256][64]
    const float* __restrict__ b_emb,  // [64]
    const float* __restrict__ convW,  // [3][64][64]
    const float* __restrict__ convB,  // [3][64]
    const float* __restrict__ W_out,  // [64][128]
    const float* __restrict__ b_out,  // [128]
    float* __restrict__ out)          // [2048][256][128]
{
    extern __shared__ float smem[];
    float* Xs    = smem;                    // 256*68  = 17408 f
    float* Hs    = Xs + NODES * XS;         // 256*68  = 17408 f
    float* Wl    = Hs + NODES * XS;         // 64*144  =  9216 f
    float* dinvS = Wl + 64 * WSTR;          // 256 f
    float* red   = dinvS + NODES;           // 8*64 = 512 f
    float* initS = red + 512;               // 64 f

    const int tid  = threadIdx.x;
    const int lane = tid & 31;
    const int wv   = tid >> 5;              // wave id 0..15
    const int g    = blockIdx.x;            // graph id

    // ---- degree^{-1/2} table (deg = self + #band neighbors) ----
    if (tid < NODES) {
        int n = tid;
        float deg = 1.0f + (n >= 1) + (n >= 2) + (n <= NODES - 3) + (n <= NODES - 2);
        dinvS[n] = rsqrtf(deg);
    }

    // ---- init = emb_row @ W_emb + b_emb  (1x256 . 256x64) ----
    {
        const float* e = emb + (size_t)g * 256;
        int c = tid & 63;
        int chunk = tid >> 6;               // 0..7, 32 k-values each
        float p = 0.0f;
        int k0 = chunk * 32;
        #pragma unroll 8
        for (int k = 0; k < 32; ++k)
            p = fmaf(e[k0 + k], W_emb[(k0 + k) * 64 + c], p);
        red[chunk * 64 + c] = p;
    }
    __syncthreads();
    if (tid < 64) {
        float s = b_emb[tid];
        #pragma unroll
        for (int r = 0; r < 8; ++r) s += red[r * 64 + tid];
        initS[tid] = s;
    }
    __syncthreads();

    // ---- broadcast: X[n][:] = init for all nodes ----
    {
        v2f iv;
        iv.x = initS[lane * 2];
        iv.y = initS[lane * 2 + 1];
        #pragma unroll
        for (int i = 0; i < 16; ++i) {
            int n = wv + 16 * i;
            *(v2f*)&Xs[n * XS + lane * 2] = iv;
        }
    }
    __syncthreads();

    const int mrow = lane & 15;             // fragment row/col within tile
    const int koff = (lane >> 4) << 1;      // K sub-offset per 16x16x4 layout

    // ================= 3 GCN layers =================
    for (int l = 0; l < 3; ++l) {
        // stage W_l (64x64) into LDS
        const float* Wg = convW + l * 64 * 64;
        for (int idx = tid; idx < 64 * 64; idx += 512)
            Wl[(idx >> 6) * WSTR + (idx & 63)] = Wg[idx];
        __syncthreads();

        // ---- H = X @ W_l   (M=256,N=64,K=64), WMMA f32 16x16x4 ----
        #pragma unroll
        for (int t = 0; t < 4; ++t) {
            int tile = wv + (t << 4);       // 0..63
            int mt = tile >> 2, nt = tile & 3;
            v8f acc = {};
            const float* Arow = &Xs[(mt * 16 + mrow) * XS + koff];
            #pragma unroll
            for (int kb = 0; kb < 16; ++kb) {
                v2f a = *(const v2f*)(Arow + kb * 4);          // ds_load_b64
                int r = kb * 4 + koff;
                v2f b;
                b.x = Wl[r * WSTR + nt * 16 + mrow];
                b.y = Wl[(r + 1) * WSTR + nt * 16 + mrow];
                acc = __builtin_amdgcn_wmma_f32_16x16x4_f32(
                        false, a, false, b, (short)0, acc, false, false);
            }
            int cn = nt * 16 + mrow;
            int mbase = mt * 16 + ((lane >> 4) << 3);
            #pragma unroll
            for (int vr = 0; vr < 8; ++vr)
                Hs[(mbase + vr) * XS + cn] = acc[vr];
        }
        __syncthreads();

        // ---- X = relu( P @ H + b_l )  (pentadiagonal, wave-uniform rows) ----
        {
            int f = lane * 2;
            float b0 = convB[l * 64 + f];
            float b1 = convB[l * 64 + f + 1];
            #pragma unroll 4
            for (int i = 0; i < 16; ++i) {
                int n = wv + 16 * i;
                float dn = dinvS[n];
                float a0 = b0, a1 = b1;
                #pragma unroll
                for (int d = -2; d <= 2; ++d) {
                    int j = n + d;                 // uniform across the wave
                    if (j >= 0 && j < NODES) {
                        float wgt = dn * dinvS[j];
                        v2f h = *(const v2f*)&Hs[j * XS + f];
                        a0 = fmaf(wgt, h.x, a0);
                        a1 = fmaf(wgt, h.y, a1);
                    }
                }
                v2f xv;
                xv.x = fmaxf(a0, 0.0f);
                xv.y = fmaxf(a1, 0.0f);
                *(v2f*)&Xs[n * XS + f] = xv;
            }
        }
        __syncthreads();
    }

    // ================= output GEMM: out = X @ W_out + b_out =================
    for (int idx = tid; idx < 64 * 128; idx += 512)
        Wl[(idx >> 7) * WSTR + (idx & 127)] = W_out[idx];
    __syncthreads();

    float* og = out + (size_t)g * NODES * OUTD;
    #pragma unroll
    for (int t = 0; t < 8; ++t) {
        int tile = wv + (t << 4);               // 0..127
        int mt = tile >> 3, nt = tile & 7;
        v8f acc = {};
        const float* Arow = &Xs[(mt * 16 + mrow) * XS + koff];
        #pragma unroll
        for (int kb = 0; kb < 16; ++kb) {
            v2f a = *(const v2f*)(Arow + kb * 4);
            int r = kb * 4 + koff;
            v2f b;
            b.x = Wl[r * WSTR + nt * 16 + mrow];
            b.y = Wl[(r + 1) * WSTR + nt * 16 + mrow];
            acc = __builtin_amdgcn_wmma_f32_16x16x4_f32(
                    false, a, false, b, (short)0, acc, false, false);
        }
        int cn = nt * 16 + mrow;
        float bo = b_out[cn];
        int mbase = mt * 16 + ((lane >> 4) << 3);
        #pragma unroll
        for (int vr = 0; vr < 8; ++vr)
            og[(size_t)(mbase + vr) * OUTD + cn] = acc[vr] + bo;
    }
}

extern "C" void kernel_launch(void* const* d_in, const int* in_sizes, int n_in,
                              void* d_out, int out_size, void* d_ws, size_t ws_size,
                              hipStream_t stream) {
    const float* emb   = (const float*)d_in[0];
    const float* W_emb = (const float*)d_in[1];
    const float* b_emb = (const float*)d_in[2];
    const float* convW = (const float*)d_in[3];
    const float* convB = (const float*)d_in[4];
    const float* W_out = (const float*)d_in[5];
    const float* b_out = (const float*)d_in[6];
    float* out = (float*)d_out;

    // dynamic LDS: X(256*68) + H(256*68) + W(64*144) + dinv(256) + red(512) + init(64)
    const size_t smem_floats = (size_t)NODES * XS * 2 + 64 * WSTR + NODES + 512 + 64;
    const size_t smem_bytes = smem_floats * sizeof(float);

    gcn_fused_kernel<<<2048, 512, smem_bytes, stream>>>(
        emb, W_emb, b_emb, convW, convB, W_out, b_out, out);
}